// RepulsionLoss_40072044872120
// MI455X (gfx1250) — compile-verified
//
#include <hip/hip_runtime.h>
#include <hip/hip_bf16.h>
#include <float.h>

// ---------------- constants from the reference ----------------
#define N_PTS   4096
#define ALPHA_C 1.0f
#define RADIUS_C 0.07f
#define H_C     0.03f
#define EPS_C   1e-12f
#define INV_H2  (1.0f / (H_C * H_C))
#define N_TERMS (8.0f * 4096.0f * 4.0f)   // mean denominator: B*N*(NN_SIZE-1)

typedef __attribute__((ext_vector_type(2))) float v2f;
typedef __attribute__((ext_vector_type(8))) float v8f;
typedef __attribute__((ext_vector_type(4))) int   v4i;

// CDNA5 async global->LDS path (ASYNCcnt), guarded so compile never breaks.
#if __has_builtin(__builtin_amdgcn_global_load_async_to_lds_b128) && \
    __has_builtin(__builtin_amdgcn_s_wait_asynccnt)
#define HAVE_ASYNC_LDS 1
typedef __attribute__((address_space(1))) v4i as1_v4i;   // global int4
typedef __attribute__((address_space(3))) v4i as3_v4i;   // LDS int4
#else
#define HAVE_ASYNC_LDS 0
#endif

// ---- dynamic LDS layout (bytes) ----
#define RED_OFF   0
#define PT_OFF    64                          // float4 pts[4096]  (64KB)
#define SQ_OFF    (PT_OFF + N_PTS * 16)       // float  sq[4096]   (16KB)
#define STAGE_OFF (SQ_OFF + N_PTS * 4)        // float  stage[12288] (48KB, async only)
#if HAVE_ASYNC_LDS
#define SHMEM_BYTES (STAGE_OFF + N_PTS * 12)
#else
#define SHMEM_BYTES STAGE_OFF
#endif

#define LESSVI(av, aj, bv, bj) (((av) < (bv)) || (((av) == (bv)) && ((aj) < (bj))))

// Branchless insert, plain strict-less (stable for ascending-j insertion order).
__device__ __forceinline__ void ins5_fast(float v, int j,
                                          float& v0, float& v1, float& v2, float& v3, float& v4,
                                          int& j0, int& j1, int& j2, int& j3, int& j4) {
  const bool l0 = v < v0;
  const bool l1 = v < v1;
  const bool l2 = v < v2;
  const bool l3 = v < v3;
  const bool l4 = v < v4;
  v4 = l3 ? v3 : (l4 ? v : v4);  j4 = l3 ? j3 : (l4 ? j : j4);
  v3 = l2 ? v2 : (l3 ? v : v3);  j3 = l2 ? j2 : (l3 ? j : j3);
  v2 = l1 ? v1 : (l2 ? v : v2);  j2 = l1 ? j1 : (l2 ? j : j2);
  v1 = l0 ? v0 : (l1 ? v : v1);  j1 = l0 ? j0 : (l1 ? j : j1);
  v0 = l0 ? v  : v0;             j0 = l0 ? j  : j0;
}

// Branchless insert with lexicographic (value,index) compare — for the lane merge,
// where index order interleaves and jax top_k tie-breaking must be reproduced.
__device__ __forceinline__ void ins5_lex(float v, int j,
                                         float& v0, float& v1, float& v2, float& v3, float& v4,
                                         int& j0, int& j1, int& j2, int& j3, int& j4) {
  const bool l0 = LESSVI(v, j, v0, j0);
  const bool l1 = LESSVI(v, j, v1, j1);
  const bool l2 = LESSVI(v, j, v2, j2);
  const bool l3 = LESSVI(v, j, v3, j3);
  const bool l4 = LESSVI(v, j, v4, j4);
  v4 = l3 ? v3 : (l4 ? v : v4);  j4 = l3 ? j3 : (l4 ? j : j4);
  v3 = l2 ? v2 : (l3 ? v : v3);  j3 = l2 ? j2 : (l3 ? j : j3);
  v2 = l1 ? v1 : (l2 ? v : v2);  j2 = l1 ? j1 : (l2 ? j : j2);
  v1 = l0 ? v0 : (l1 ? v : v1);  j1 = l0 ? j0 : (l1 ? j : j1);
  v0 = l0 ? v  : v0;             j0 = l0 ? j  : j0;
}

__global__ void repulsion_stage1(const float* __restrict__ pred,
                                 float* __restrict__ partials) {
  extern __shared__ unsigned char smem_raw[];
  float*  s_red = (float*)(smem_raw + RED_OFF);   // 8 floats
  float4* s_pt  = (float4*)(smem_raw + PT_OFF);   // x,y,z,|p|^2
  float*  s_sq  = (float*)(smem_raw + SQ_OFF);    // |p|^2 (vector-loadable)

  const int tid   = threadIdx.x;
  const int batch = blockIdx.y;
  const float* pb = pred + (size_t)batch * N_PTS * 3;

#if HAVE_ASYNC_LDS
  {
    float* stage = (float*)(smem_raw + STAGE_OFF);          // raw 12288 floats
    const float4* src = (const float4*)pb;                  // 3072 x 16B, aligned
    for (int t = tid; t < (N_PTS * 3) / 4; t += 256) {
      __builtin_amdgcn_global_load_async_to_lds_b128(
          (as1_v4i*)(src + t), (as3_v4i*)(stage + 4 * t), 0, 0);
    }
    __builtin_amdgcn_s_wait_asynccnt(0);
    __syncthreads();
    for (int i = tid; i < N_PTS; i += 256) {
      const float x = stage[3 * i + 0];
      const float y = stage[3 * i + 1];
      const float z = stage[3 * i + 2];
      const float sq = x * x + y * y + z * z;
      s_pt[i] = make_float4(x, y, z, sq);
      s_sq[i] = sq;
    }
  }
#else
  for (int i = tid; i < N_PTS; i += 256) {
    const float x = pb[3 * i + 0];
    const float y = pb[3 * i + 1];
    const float z = pb[3 * i + 2];
    const float sq = x * x + y * y + z * z;
    s_pt[i] = make_float4(x, y, z, sq);
    s_sq[i] = sq;
  }
#endif
  __syncthreads();

  const int  wave  = tid >> 5;
  const int  lane  = tid & 31;
  const int  li    = lane & 15;
  const bool hi    = lane >= 16;
  const int  hioff = hi ? 8 : 0;
  const int  qbase = blockIdx.x * 128 + wave * 16;
  const int  qi    = qbase + li;

  // B fragment (4x16, queries scaled by -2, built ONCE per wave):
  //   v0: lanes 0-15 -> B[0][n]=-2x_q ; lanes 16-31 -> B[1][n]=-2y_q
  //   v1: lanes 0-15 -> B[2][n]=-2z_q ; lanes 16-31 -> B[3][n]=0
  const float4 pq = s_pt[qi];
  v2f bfrag;
  bfrag[0] = hi ? (-2.0f * pq.y) : (-2.0f * pq.x);
  bfrag[1] = hi ? 0.0f : (-2.0f * pq.z);
  const float sqi = pq.w;

  float v0 = FLT_MAX, v1 = FLT_MAX, v2 = FLT_MAX, v3 = FLT_MAX, v4 = FLT_MAX;
  int   j0 = 0x7fffffff, j1 = 0x7fffffff, j2 = 0x7fffffff, j3 = 0x7fffffff, j4 = 0x7fffffff;

#pragma unroll 2
  for (int jb = 0; jb < N_PTS; jb += 16) {
    // A fragment (16x4, raw candidate coords): lane L holds row m=L%16.
    const float4 pj = s_pt[jb + li];
    v2f afrag;
    afrag[0] = hi ? pj.z : pj.x;
    afrag[1] = hi ? 0.0f : pj.y;

    // C[m][n] = |p_j|^2 + |p_i|^2 via two broadcast b128 loads from s_sq.
    const float4 sa = *(const float4*)(s_sq + jb + hioff);
    const float4 sb = *(const float4*)(s_sq + jb + hioff + 4);
    v8f cfrag;
    cfrag[0] = sa.x + sqi; cfrag[1] = sa.y + sqi;
    cfrag[2] = sa.z + sqi; cfrag[3] = sa.w + sqi;
    cfrag[4] = sb.x + sqi; cfrag[5] = sb.y + sqi;
    cfrag[6] = sb.z + sqi; cfrag[7] = sb.w + sqi;

    // D[m][n] = -2 p_j . p_i + |p_j|^2 + |p_i|^2   (f32 WMMA, K=4, exact)
    const v8f d = __builtin_amdgcn_wmma_f32_16x16x4_f32(
        false, afrag, false, bfrag, (short)0, cfrag, false, false);

    // Early-out: does any of this lane's 8 candidates beat the current 5th-best?
    const float m01 = fminf(d[0], d[1]);
    const float m23 = fminf(d[2], d[3]);
    const float m45 = fminf(d[4], d[5]);
    const float m67 = fminf(d[6], d[7]);
    const float mall = fminf(fminf(m01, m23), fminf(m45, m67));
    if (mall <= v4) {
#pragma unroll
      for (int k = 0; k < 8; ++k)
        ins5_fast(d[k], jb + hioff + k, v0, v1, v2, v3, v4, j0, j1, j2, j3, j4);
    }
  }

  // Merge the two half-lists for query li across lanes L and L+16.
  // Snapshot partner's list BEFORE mutating ours (lockstep wave32).
  const float pv0 = __shfl_xor(v0, 16, 32), pv1 = __shfl_xor(v1, 16, 32),
              pv2 = __shfl_xor(v2, 16, 32), pv3 = __shfl_xor(v3, 16, 32),
              pv4 = __shfl_xor(v4, 16, 32);
  const int   pj0 = __shfl_xor(j0, 16, 32), pj1 = __shfl_xor(j1, 16, 32),
              pj2 = __shfl_xor(j2, 16, 32), pj3 = __shfl_xor(j3, 16, 32),
              pj4 = __shfl_xor(j4, 16, 32);
  ins5_lex(pv0, pj0, v0, v1, v2, v3, v4, j0, j1, j2, j3, j4);
  ins5_lex(pv1, pj1, v0, v1, v2, v3, v4, j0, j1, j2, j3, j4);
  ins5_lex(pv2, pj2, v0, v1, v2, v3, v4, j0, j1, j2, j3, j4);
  ins5_lex(pv3, pj3, v0, v1, v2, v3, v4, j0, j1, j2, j3, j4);
  ins5_lex(pv4, pj4, v0, v1, v2, v3, v4, j0, j1, j2, j3, j4);

  // Lanes 0-15 hold the top-5 for their query; drop slot 0 (self), accumulate
  // the loss with dist recomputed exactly as the reference (from diff).
  float acc = 0.0f;
  if (!hi) {
    const int nidx[4] = { j1, j2, j3, j4 };
#pragma unroll
    for (int k = 0; k < 4; ++k) {
      const float4 pn = s_pt[nidx[k]];
      const float dx = pn.x - pq.x;
      const float dy = pn.y - pq.y;
      const float dz = pn.z - pq.z;
      float d2v = dx * dx + dy * dy + dz * dz;
      d2v = fmaxf(d2v, EPS_C);
      const float dist = sqrtf(d2v);
      const float w = expf(-d2v * INV_H2);
      acc += (RADIUS_C - dist) * w;
    }
  }

  for (int off = 16; off >= 1; off >>= 1) acc += __shfl_xor(acc, off, 32);
  if (lane == 0) s_red[wave] = acc;
  __syncthreads();
  if (tid == 0) {
    float t = 0.0f;
#pragma unroll
    for (int w = 0; w < 8; ++w) t += s_red[w];
    partials[blockIdx.y * gridDim.x + blockIdx.x] = t;
  }
}

// Stage 2: deterministic reduction of the 256 block partials -> scalar mean.
__global__ void repulsion_stage2(const float* __restrict__ partials,
                                 float* __restrict__ out) {
  __shared__ float s_red[8];
  const int tid = threadIdx.x;
  float v = partials[tid];
  for (int off = 16; off >= 1; off >>= 1) v += __shfl_xor(v, off, 32);
  if ((tid & 31) == 0) s_red[tid >> 5] = v;
  __syncthreads();
  if (tid == 0) {
    float t = 0.0f;
#pragma unroll
    for (int w = 0; w < 8; ++w) t += s_red[w];
    out[0] = ALPHA_C * (t / N_TERMS);
  }
}

extern "C" void kernel_launch(void* const* d_in, const int* in_sizes, int n_in,
                              void* d_out, int out_size, void* d_ws, size_t ws_size,
                              hipStream_t stream) {
  (void)in_sizes; (void)n_in; (void)out_size; (void)ws_size;
  const float* pred = (const float*)d_in[0];     // (8, 4096, 3) f32
  float* partials = (float*)d_ws;                // 256 floats of scratch
  float* out = (float*)d_out;                    // scalar f32

  dim3 grid(32, 8, 1);     // 32 row-blocks of 128 queries x 8 batches
  dim3 block(256, 1, 1);   // 8 wave32 = 8 x 16 queries

  repulsion_stage1<<<grid, block, (size_t)SHMEM_BYTES, stream>>>(pred, partials);
  repulsion_stage2<<<dim3(1, 1, 1), dim3(256, 1, 1), 0, stream>>>(partials, out);
}